// MambaBlock_23089744183592
// MI455X (gfx1250) — compile-verified
//
#include <hip/hip_runtime.h>
#include <hip/hip_bf16.h>
#include <math.h>

// ---------------- problem constants ----------------
#define DMODEL 768
#define DINNER 1536
#define DSTATE 16
#define DCONV  4
#define BB     2
#define LL     2048
#define MROWS  (BB * LL)              // 4096
#define XDBL_N (DINNER + 2 * DSTATE)  // 1568

// chunked scan
#define NCHUNK 32
#define CLEN   (LL / NCHUNK)          // 64
#define CHST   (BB * DSTATE * DINNER) // per-chunk stride in P/Q/Hinit = 49152

typedef __attribute__((ext_vector_type(2))) float v2f;
typedef __attribute__((ext_vector_type(8))) float v8f;

// ---------------- LayerNorm ----------------
__global__ __launch_bounds__(256) void ln_kernel(
    const float* __restrict__ x, const float* __restrict__ gamma,
    const float* __restrict__ beta, float* __restrict__ xn)
{
    const int row = blockIdx.x;
    const int t   = threadIdx.x;
    const float* xr = x + (size_t)row * DMODEL;

    float v[3];
    float s = 0.f, s2 = 0.f;
#pragma unroll
    for (int i = 0; i < 3; ++i) {
        v[i] = xr[t + i * 256];
        s  += v[i];
        s2 += v[i] * v[i];
    }
    __shared__ float rs[256];
    __shared__ float rq[256];
    rs[t] = s; rq[t] = s2;
    __syncthreads();
    for (int off = 128; off > 0; off >>= 1) {
        if (t < off) { rs[t] += rs[t + off]; rq[t] += rq[t + off]; }
        __syncthreads();
    }
    const float mean = rs[0] * (1.0f / DMODEL);
    const float var  = rq[0] * (1.0f / DMODEL) - mean * mean;
    const float inv  = rsqrtf(var + 1e-5f);
    float* outr = xn + (size_t)row * DMODEL;
#pragma unroll
    for (int i = 0; i < 3; ++i) {
        const int c = t + i * 256;
        outr[c] = (v[i] - mean) * inv * gamma[c] + beta[c];
    }
}

// ---------------- generic fp32 WMMA GEMM ----------------
// C[M x N] = A[M x K] @ B[K x N]  (+ epilogue)
//   mode 0: none
//   mode 1: softplus(val + bias[col])       (fused dt_proj)
//   mode 2: val + addend[row*ldc + col]     (fused residual add)
#define BLK_M 128
#define BLK_N 64
#define BLK_K 32
#define LDA_S 36   // padded LDS stride for A tile (mult of 4, conflict-free)
#define LDB_S 68   // padded LDS stride for B tile

__global__ __launch_bounds__(256) void gemm_kernel(
    const float* __restrict__ A, const float* __restrict__ B,
    float* __restrict__ C, int M, int N, int K,
    int lda, int ldb, int ldc,
    int mode, const float* __restrict__ bias, const float* __restrict__ addend)
{
    __shared__ float As[BLK_M * LDA_S];   // 18 KB
    __shared__ float Bs[BLK_K * LDB_S];   //  8.5 KB

    const int tid  = threadIdx.x;
    const int lane = tid & 31;
    const int wave = tid >> 5;           // 0..7
    const int wm   = wave >> 1;          // 0..3 -> M offset wm*32
    const int wn   = wave & 1;           // 0..1 -> N offset wn*32
    const int bm   = blockIdx.x * BLK_M;
    const int bn   = blockIdx.y * BLK_N;

    const int l15   = lane & 15;
    const int khalf = (lane >> 4) * 2;   // lanes 0-15 -> K {0,1}; 16-31 -> K {2,3}

    v8f acc[2][2] = {};

    for (int k0 = 0; k0 < K; k0 += BLK_K) {
        // stage A tile 128x32 (4 x float4 per thread)
#pragma unroll
        for (int i = 0; i < 4; ++i) {
            const int t  = tid + i * 256;       // 0..1023 float4 slots
            const int r  = t >> 3;              // 8 float4 per row
            const int c4 = (t & 7) << 2;
            const float4 av = *(const float4*)(A + (size_t)(bm + r) * lda + k0 + c4);
            *(float4*)(As + r * LDA_S + c4) = av;
        }
        // stage B tile 32x64 (2 x float4 per thread), guarded in N
#pragma unroll
        for (int i = 0; i < 2; ++i) {
            const int t  = tid + i * 256;       // 0..511 float4 slots
            const int r  = t >> 4;              // 16 float4 per row
            const int c4 = (t & 15) << 2;
            const int gn = bn + c4;
            float4 bv;
            if (gn + 3 < N) {
                bv = *(const float4*)(B + (size_t)(k0 + r) * ldb + gn);
            } else {
                float tmp[4] = {0.f, 0.f, 0.f, 0.f};
                for (int q = 0; q < 4; ++q)
                    if (gn + q < N) tmp[q] = B[(size_t)(k0 + r) * ldb + gn + q];
                bv = make_float4(tmp[0], tmp[1], tmp[2], tmp[3]);
            }
            *(float4*)(Bs + r * LDB_S + c4) = bv;
        }
        // speculative prefetch of next K-tile into the cache hierarchy
        if (k0 + BLK_K < K) {
            const int r = tid & 127;
            __builtin_prefetch(A + (size_t)(bm + r) * lda + (k0 + BLK_K) + ((tid >> 7) << 4), 0, 1);
            if (tid < 128) {
                const int rr = tid >> 2;           // 0..31
                const int cc = (tid & 3) << 4;     // 0,16,32,48
                __builtin_prefetch(B + (size_t)(k0 + BLK_K + rr) * ldb + bn + cc, 0, 1);
            }
        }
        __syncthreads();

#pragma unroll
        for (int kk = 0; kk < BLK_K; kk += 4) {
            v2f afrag[2], bfrag[2];
#pragma unroll
            for (int i = 0; i < 2; ++i) {
                const int m = wm * 32 + i * 16 + l15;
                const float* ap = As + m * LDA_S + kk + khalf;
                afrag[i].x = ap[0];
                afrag[i].y = ap[1];
            }
#pragma unroll
            for (int j = 0; j < 2; ++j) {
                const int n = wn * 32 + j * 16 + l15;
                const float* bp = Bs + (kk + khalf) * LDB_S + n;
                bfrag[j].x = bp[0];
                bfrag[j].y = bp[LDB_S];
            }
#pragma unroll
            for (int i = 0; i < 2; ++i)
#pragma unroll
                for (int j = 0; j < 2; ++j)
                    acc[i][j] = __builtin_amdgcn_wmma_f32_16x16x4_f32(
                        false, afrag[i], false, bfrag[j],
                        (short)0, acc[i][j], false, false);
        }
        __syncthreads();
    }

    // store: v8f layout -> lanes 0-15: M=vgpr, N=lane; lanes 16-31: M=vgpr+8, N=lane-16
    const int rbase = bm + wm * 32 + ((lane >> 4) ? 8 : 0);
#pragma unroll
    for (int i = 0; i < 2; ++i) {
#pragma unroll
        for (int j = 0; j < 2; ++j) {
            const int col = bn + wn * 32 + j * 16 + l15;
            if (col >= N) continue;
#pragma unroll
            for (int v = 0; v < 8; ++v) {
                const int row = rbase + i * 16 + v;
                float val = acc[i][j][v];
                if (mode == 1) {
                    const float sarg = val + bias[col];
                    val = (sarg > 20.f) ? sarg : __logf(1.f + __expf(sarg));
                } else if (mode == 2) {
                    val += addend[(size_t)row * ldc + col];
                }
                C[(size_t)row * ldc + col] = val;
            }
        }
    }
}

// ---------------- depthwise causal conv1d + bias + SiLU ----------------
// x_br = xz[:, :, 0:DINNER]  (row stride 2*DINNER)
__global__ __launch_bounds__(256) void conv_silu_kernel(
    const float* __restrict__ xz, const float* __restrict__ w,
    const float* __restrict__ bias, float* __restrict__ xc)
{
    const int idx = blockIdx.x * 256 + threadIdx.x;     // over B*L*DINNER
    const int d = idx % DINNER;
    const int l = (idx / DINNER) % LL;
    const int b = idx / (DINNER * LL);

    float acc = bias[d];
#pragma unroll
    for (int j = 0; j < DCONV; ++j) {
        const int t = l - (DCONV - 1) + j;
        if (t >= 0)
            acc += xz[((size_t)(b * LL + t)) * (2 * DINNER) + d] * w[d * DCONV + j];
    }
    const float sig = 1.f / (1.f + __expf(-acc));
    xc[idx] = acc * sig;     // SiLU
}

// ---------------- chunked selective scan ----------------
// h_t = dA_t * h_{t-1} + dBx_t is a linear recurrence per (b,d,n).
// Phase 1: per chunk compute composed transition (p = prod dA, q = state from 0).
// Phase 2: sequential prefix over the NCHUNK chunks -> Hinit per chunk.
// Phase 3: replay each chunk from Hinit, produce y and SiLU(z)-gated output.
// P/Q/Hinit layout: [c][b][n][d]  (d fastest -> coalesced everywhere)

__global__ __launch_bounds__(256) void scan_phase1_kernel(
    const float* __restrict__ delta, const float* __restrict__ xc,
    const float* __restrict__ xdbl,  const float* __restrict__ A_log,
    float* __restrict__ P, float* __restrict__ Q)
{
    const int idx = blockIdx.x * 256 + threadIdx.x;     // (c*BB + b)*DINNER + d
    const int d  = idx % DINNER;
    const int cb = idx / DINNER;                        // c*BB + b
    const int b  = cb % BB;
    const int c  = cb / BB;

    float Arow[DSTATE];
#pragma unroll
    for (int n = 0; n < DSTATE; ++n)
        Arow[n] = -__expf(A_log[d * DSTATE + n]);

    float p[DSTATE], q[DSTATE];
#pragma unroll
    for (int n = 0; n < DSTATE; ++n) { p[n] = 1.f; q[n] = 0.f; }

    const int t0 = c * CLEN;
    for (int i = 0; i < CLEN; ++i) {
        const size_t row = (size_t)(b * LL + t0 + i);
        const float dt = delta[row * DINNER + d];
        const float xt = xc[row * DINNER + d];
        const float* bptr = xdbl + row * XDBL_N + DINNER;   // B_t[0..15]
        const float dx = dt * xt;
#pragma unroll
        for (int n = 0; n < DSTATE; ++n) {
            const float dA = __expf(dt * Arow[n]);
            q[n] = dA * q[n] + dx * bptr[n];
            p[n] *= dA;
        }
    }
    float* Pb = P + (size_t)cb * DSTATE * DINNER;
    float* Qb = Q + (size_t)cb * DSTATE * DINNER;
#pragma unroll
    for (int n = 0; n < DSTATE; ++n) {
        Pb[n * DINNER + d] = p[n];
        Qb[n * DINNER + d] = q[n];
    }
}

__global__ __launch_bounds__(256) void scan_phase2_kernel(
    const float* __restrict__ P, const float* __restrict__ Q,
    float* __restrict__ Hinit)
{
    const int j = blockIdx.x * 256 + threadIdx.x;   // (b*DSTATE + n)*DINNER + d
    float h = 0.f;
#pragma unroll 4
    for (int c = 0; c < NCHUNK; ++c) {
        Hinit[(size_t)c * CHST + j] = h;
        h = P[(size_t)c * CHST + j] * h + Q[(size_t)c * CHST + j];
    }
}

__global__ __launch_bounds__(256) void scan_phase3_kernel(
    const float* __restrict__ delta, const float* __restrict__ xc,
    const float* __restrict__ xdbl,  const float* __restrict__ xz,
    const float* __restrict__ A_log, const float* __restrict__ Dp_,
    const float* __restrict__ Hinit, float* __restrict__ u)
{
    const int idx = blockIdx.x * 256 + threadIdx.x;     // (c*BB + b)*DINNER + d
    const int d  = idx % DINNER;
    const int cb = idx / DINNER;
    const int b  = cb % BB;
    const int c  = cb / BB;

    float Arow[DSTATE];
#pragma unroll
    for (int n = 0; n < DSTATE; ++n)
        Arow[n] = -__expf(A_log[d * DSTATE + n]);

    float h[DSTATE];
    const float* Hb = Hinit + (size_t)cb * DSTATE * DINNER;
#pragma unroll
    for (int n = 0; n < DSTATE; ++n)
        h[n] = Hb[n * DINNER + d];

    const float Dp = Dp_[d];
    const int t0 = c * CLEN;
    for (int i = 0; i < CLEN; ++i) {
        const size_t row = (size_t)(b * LL + t0 + i);
        const float dt = delta[row * DINNER + d];
        const float xt = xc[row * DINNER + d];
        const float* bc = xdbl + row * XDBL_N;      // B at +DINNER, C at +DINNER+DSTATE
        const float dx = dt * xt;
        float y = 0.f;
#pragma unroll
        for (int n = 0; n < DSTATE; ++n) {
            const float dA = __expf(dt * Arow[n]);
            h[n] = dA * h[n] + dx * bc[DINNER + n];
            y += h[n] * bc[DINNER + DSTATE + n];
        }
        y += Dp * xt;
        const float z = xz[row * (2 * DINNER) + DINNER + d];
        const float sz = z / (1.f + __expf(-z));    // SiLU(z)
        u[row * DINNER + d] = y * sz;
    }
}

// ---------------- launcher ----------------
extern "C" void kernel_launch(void* const* d_in, const int* in_sizes, int n_in,
                              void* d_out, int out_size, void* d_ws, size_t ws_size,
                              hipStream_t stream) {
    const float* x          = (const float*)d_in[0];
    const float* in_proj_w  = (const float*)d_in[1];
    const float* conv_w     = (const float*)d_in[2];
    const float* conv_b     = (const float*)d_in[3];
    const float* x_proj_w   = (const float*)d_in[4];
    const float* dt_proj_w  = (const float*)d_in[5];
    const float* dt_proj_b  = (const float*)d_in[6];
    const float* A_log      = (const float*)d_in[7];
    const float* D_param    = (const float*)d_in[8];
    const float* out_proj_w = (const float*)d_in[9];
    const float* ln_gamma   = (const float*)d_in[10];
    const float* ln_beta    = (const float*)d_in[11];
    float* out = (float*)d_out;

    float* ws    = (float*)d_ws;
    float* xn    = ws;                                      // 4096*768
    float* xz    = xn    + (size_t)MROWS * DMODEL;          // 4096*3072
    float* xc    = xz    + (size_t)MROWS * 2 * DINNER;      // 4096*1536
    float* xdbl  = xc    + (size_t)MROWS * DINNER;          // 4096*1568
    float* delta = xdbl  + (size_t)MROWS * XDBL_N;          // 4096*1536
    float* u     = delta + (size_t)MROWS * DINNER;          // 4096*1536
    float* Pbuf  = u     + (size_t)MROWS * DINNER;          // NCHUNK*CHST
    float* Qbuf  = Pbuf  + (size_t)NCHUNK * CHST;           // NCHUNK*CHST
    float* Hin   = Qbuf  + (size_t)NCHUNK * CHST;           // NCHUNK*CHST

    // 1. LayerNorm
    ln_kernel<<<MROWS, 256, 0, stream>>>(x, ln_gamma, ln_beta, xn);

    // 2. xz = xn @ in_proj_w            (4096x768 @ 768x3072)
    gemm_kernel<<<dim3(MROWS / BLK_M, (2 * DINNER) / BLK_N), 256, 0, stream>>>(
        xn, in_proj_w, xz, MROWS, 2 * DINNER, DMODEL,
        DMODEL, 2 * DINNER, 2 * DINNER, 0, nullptr, nullptr);

    // 3. depthwise causal conv + SiLU
    conv_silu_kernel<<<(MROWS * DINNER) / 256, 256, 0, stream>>>(
        xz, conv_w, conv_b, xc);

    // 4. x_dbl = xc @ x_proj_w          (4096x1536 @ 1536x1568)
    gemm_kernel<<<dim3(MROWS / BLK_M, (XDBL_N + BLK_N - 1) / BLK_N), 256, 0, stream>>>(
        xc, x_proj_w, xdbl, MROWS, XDBL_N, DINNER,
        DINNER, XDBL_N, XDBL_N, 0, nullptr, nullptr);

    // 5. delta = softplus(x_dbl[:, :1536] @ dt_proj_w + dt_b)   (lda = 1568)
    gemm_kernel<<<dim3(MROWS / BLK_M, DINNER / BLK_N), 256, 0, stream>>>(
        xdbl, dt_proj_w, delta, MROWS, DINNER, DINNER,
        XDBL_N, DINNER, DINNER, 1, dt_proj_b, nullptr);

    // 6. chunked selective scan + gating with SiLU(z) -> u
    scan_phase1_kernel<<<(NCHUNK * BB * DINNER) / 256, 256, 0, stream>>>(
        delta, xc, xdbl, A_log, Pbuf, Qbuf);
    scan_phase2_kernel<<<CHST / 256, 256, 0, stream>>>(Pbuf, Qbuf, Hin);
    scan_phase3_kernel<<<(NCHUNK * BB * DINNER) / 256, 256, 0, stream>>>(
        delta, xc, xdbl, xz, A_log, D_param, Hin, u);

    // 7. out = u @ out_proj_w + residual(x)    (4096x1536 @ 1536x768)
    gemm_kernel<<<dim3(MROWS / BLK_M, DMODEL / BLK_N), 256, 0, stream>>>(
        u, out_proj_w, out, MROWS, DMODEL, DINNER,
        DINNER, DMODEL, DMODEL, 2, nullptr, x);
}